// QuantumHeadAngle_54692113547572
// MI455X (gfx1250) — compile-verified
//
#include <hip/hip_runtime.h>

typedef __attribute__((ext_vector_type(2))) float v2f;
typedef __attribute__((ext_vector_type(8))) float v8f;

#define N_QUBITS 12
#define DIM 4096
#define N_LAYERS 2
#define NUM_CLASSES 10
#define BATCH 1024

// ---------------------------------------------------------------------------
// Kernel 1: build the six fused 16x16 complex unitaries
//   G[l][g] = Rot(l,4g) (x) Rot(l,4g+1) (x) Rot(l,4g+2) (x) Rot(l,4g+3)
// ws layout: re: ws[(l*3+g)*256 + m*16 + k], im: ws[1536 + ...]
// ---------------------------------------------------------------------------
__global__ __launch_bounds__(256) void build_gates_kernel(
    const float* __restrict__ qp, float* __restrict__ ws) {
  int t = threadIdx.x;            // 0..255 -> (m,k) entry of the 16x16 gate
  int m = t >> 4, kcol = t & 15;
  for (int lg = 0; lg < 6; ++lg) {
    int l = lg / 3, g = lg % 3;
    float pr = 1.f, pi = 0.f;     // running complex product
    #pragma unroll
    for (int k = 0; k < 4; ++k) {
      int q = 4 * g + k;
      float phi = qp[(l * N_QUBITS + q) * 3 + 0];
      float th  = qp[(l * N_QUBITS + q) * 3 + 1];
      float om  = qp[(l * N_QUBITS + q) * 3 + 2];
      float ct = __cosf(0.5f * th), st = __sinf(0.5f * th);
      int ak = (m >> (3 - k)) & 1;   // row bit (qubit 4g+k is MSB of nibble)
      int bk = (kcol >> (3 - k)) & 1;
      float er, ei;
      if (ak == 0 && bk == 0) {       // exp(-i(phi+om)/2)*ct
        float an = -0.5f * (phi + om); er = __cosf(an) * ct; ei = __sinf(an) * ct;
      } else if (ak == 0 && bk == 1) {// -exp(+i(phi-om)/2)*st
        float an =  0.5f * (phi - om); er = -__cosf(an) * st; ei = -__sinf(an) * st;
      } else if (ak == 1 && bk == 0) {// exp(-i(phi-om)/2)*st
        float an = -0.5f * (phi - om); er = __cosf(an) * st; ei = __sinf(an) * st;
      } else {                        // exp(+i(phi+om)/2)*ct
        float an =  0.5f * (phi + om); er = __cosf(an) * ct; ei = __sinf(an) * ct;
      }
      float nr = pr * er - pi * ei;
      float ni = pr * ei + pi * er;
      pr = nr; pi = ni;
    }
    ws[lg * 256 + t]        = pr;
    ws[1536 + lg * 256 + t] = pi;
  }
}

// ---------------------------------------------------------------------------
// Apply a fused 16x16 complex unitary along one axis of the (16,16,16) state
// held in LDS. Element address = k*ks + n*ns + t*ts; 16 column-tiles, each of
// the 8 waves owns tiles {2w, 2w+1}. Complex GEMM = 4 real 16x16x16 GEMMs,
// each = 4 chained V_WMMA_F32_16X16X4_F32.
// A-format (ISA 7.12.2): elem(m = lane&15, K = j + 2*(lane>>4)) per VGPR j.
// B-format assumed mirror: elem(K = j + 2*(lane>>4), n = lane&15).
// C/D: VGPR v -> (M = v + 8*(lane>>4), N = lane&15).
// ---------------------------------------------------------------------------
__device__ __forceinline__ void apply_gate16(
    float* sre, float* sim,
    const float* __restrict__ gwr, const float* __restrict__ gwi,
    int ks, int ns, int ts, int lane, int wave) {
  int lo = lane & 15, hi = lane >> 4;

  v2f ur[4], ui[4], nui[4];
  #pragma unroll
  for (int kk = 0; kk < 4; ++kk) {
    #pragma unroll
    for (int j = 0; j < 2; ++j) {
      int k = kk * 4 + j + 2 * hi;
      float r = gwr[lo * 16 + k];
      float im = gwi[lo * 16 + k];
      ur[kk][j] = r;
      ui[kk][j] = im;
      nui[kk][j] = -im;          // f32 WMMA NEG supports only CNeg -> VALU negate
    }
  }

  #pragma unroll
  for (int tt = 0; tt < 2; ++tt) {
    int t = 2 * wave + tt;
    v2f prm[4], pim[4];
    #pragma unroll
    for (int kk = 0; kk < 4; ++kk) {
      #pragma unroll
      for (int j = 0; j < 2; ++j) {
        int k = kk * 4 + j + 2 * hi;
        int addr = k * ks + lo * ns + t * ts;
        prm[kk][j] = sre[addr];
        pim[kk][j] = sim[addr];
      }
    }
    v8f ar = {}, ai = {};
    #pragma unroll
    for (int kk = 0; kk < 4; ++kk)   // Ur*Pr
      ar = __builtin_amdgcn_wmma_f32_16x16x4_f32(false, ur[kk],  false, prm[kk], (short)0, ar, false, false);
    #pragma unroll
    for (int kk = 0; kk < 4; ++kk)   // - Ui*Pi
      ar = __builtin_amdgcn_wmma_f32_16x16x4_f32(false, nui[kk], false, pim[kk], (short)0, ar, false, false);
    #pragma unroll
    for (int kk = 0; kk < 4; ++kk)   // Ur*Pi
      ai = __builtin_amdgcn_wmma_f32_16x16x4_f32(false, ur[kk],  false, pim[kk], (short)0, ai, false, false);
    #pragma unroll
    for (int kk = 0; kk < 4; ++kk)   // + Ui*Pr
      ai = __builtin_amdgcn_wmma_f32_16x16x4_f32(false, ui[kk],  false, prm[kk], (short)0, ai, false, false);

    #pragma unroll
    for (int v = 0; v < 8; ++v) {
      int m = v + 8 * hi;
      int addr = m * ks + lo * ns + t * ts;
      sre[addr] = ar[v];
      sim[addr] = ai[v];
    }
  }
}

// ---------------------------------------------------------------------------
// Kernel 2: one workgroup (256 threads = 8 wave32) per batch sample.
// State lives in LDS for the whole circuit.
// ---------------------------------------------------------------------------
__global__ __launch_bounds__(256) void qsim_kernel(
    const float* __restrict__ x, const float* __restrict__ W,
    const float* __restrict__ bias, const float* __restrict__ gw,
    float* __restrict__ out) {
  __shared__ float bufA_re[DIM], bufA_im[DIM];
  __shared__ float bufB_re[DIM], bufB_im[DIM];
  __shared__ float cs[N_QUBITS], sn[N_QUBITS];
  __shared__ float feats[N_QUBITS];

  int tid  = threadIdx.x;
  int lane = tid & 31, wave = tid >> 5;
  int b = blockIdx.x;

  if (tid < N_QUBITS) {
    float h = 0.5f * x[b * N_QUBITS + tid];
    cs[tid] = __cosf(h);
    sn[tid] = __sinf(h);
    feats[tid] = 0.f;
  }
  __syncthreads();

  // RX embedding on |0..0> in closed form: product state with (-i)^popcount phase
  for (int k = 0; k < 16; ++k) {
    int i = k * 256 + tid;
    float mag = 1.f;
    #pragma unroll
    for (int q = 0; q < N_QUBITS; ++q)
      mag *= ((i >> (11 - q)) & 1) ? sn[q] : cs[q];
    int ph = __popc(i) & 3;          // (-i)^ph
    float prp = (ph == 0) ? 1.f : (ph == 2) ? -1.f : 0.f;
    float pip = (ph == 1) ? -1.f : (ph == 3) ? 1.f : 0.f;
    bufA_re[i] = mag * prp;
    bufA_im[i] = mag * pip;
  }
  __syncthreads();

  float* cre = bufA_re; float* cim = bufA_im;
  float* ore = bufB_re; float* oim = bufB_im;

  for (int l = 0; l < N_LAYERS; ++l) {
    const float* gr = gw + (l * 3) * 256;
    const float* gi = gw + 1536 + (l * 3) * 256;
    // qubits 0-3 -> index bits 11:8 (stride 256)
    apply_gate16(cre, cim, gr,        gi,        256, 1, 16,  lane, wave);
    __syncthreads();
    // qubits 4-7 -> bits 7:4 (stride 16)
    apply_gate16(cre, cim, gr + 256,  gi + 256,  16,  1, 256, lane, wave);
    __syncthreads();
    // qubits 8-11 -> bits 3:0 (stride 1)
    apply_gate16(cre, cim, gr + 512,  gi + 512,  1,  16, 256, lane, wave);
    __syncthreads();

    // CNOT ring with shift r = l+1: sequential GF(2) bit updates -> scatter
    int r = (l % (N_QUBITS - 1)) + 1;
    for (int k = 0; k < 16; ++k) {
      int i = k * 256 + tid;
      int v = i;
      #pragma unroll
      for (int q = 0; q < N_QUBITS; ++q) {
        int pc = 11 - q;
        int pt = 11 - ((q + r) % N_QUBITS);
        v ^= ((v >> pc) & 1) << pt;
      }
      ore[v] = cre[i];
      oim[v] = cim[i];
    }
    __syncthreads();
    float* t0 = cre; cre = ore; ore = t0;
    float* t1 = cim; cim = oim; oim = t1;
  }

  // Per-wire <Z> expectation values
  float local[N_QUBITS];
  #pragma unroll
  for (int q = 0; q < N_QUBITS; ++q) local[q] = 0.f;
  for (int k = 0; k < 16; ++k) {
    int i = k * 256 + tid;
    float re = cre[i], im = cim[i];
    float p = re * re + im * im;
    #pragma unroll
    for (int q = 0; q < N_QUBITS; ++q)
      local[q] += ((i >> (11 - q)) & 1) ? -p : p;
  }
  #pragma unroll
  for (int q = 0; q < N_QUBITS; ++q)
    atomicAdd(&feats[q], local[q]);     // ds_add_f32
  __syncthreads();

  // Linear head: out[b,c] = feats . W[c,:] + bias[c]
  if (tid < NUM_CLASSES) {
    float acc = bias[tid];
    #pragma unroll
    for (int q = 0; q < N_QUBITS; ++q)
      acc += feats[q] * W[tid * N_QUBITS + q];
    out[b * NUM_CLASSES + tid] = acc;
  }
}

// ---------------------------------------------------------------------------
extern "C" void kernel_launch(void* const* d_in, const int* in_sizes, int n_in,
                              void* d_out, int out_size, void* d_ws, size_t ws_size,
                              hipStream_t stream) {
  const float* x  = (const float*)d_in[0];   // (1024, 12)
  const float* qp = (const float*)d_in[1];   // (2, 12, 3)
  const float* W  = (const float*)d_in[2];   // (10, 12)
  const float* bb = (const float*)d_in[3];   // (10,)
  float* out = (float*)d_out;                // (1024, 10)
  float* gw  = (float*)d_ws;                 // 3072 floats = 12 KB

  build_gates_kernel<<<1, 256, 0, stream>>>(qp, gw);
  qsim_kernel<<<BATCH, 256, 0, stream>>>(x, W, bb, gw, out);
}